// invencoder_42090679501108
// MI455X (gfx1250) — compile-verified
//
#include <hip/hip_runtime.h>
#include <math.h>

#define EPSF 1e-5f
#define SQRT3F 1.7320508075688772f

typedef float v2f __attribute__((ext_vector_type(2)));
typedef float v8f __attribute__((ext_vector_type(8)));

__device__ inline v8f v8f_zero() {
    v8f v;
#pragma unroll
    for (int i = 0; i < 8; ++i) v[i] = 0.f;
    return v;
}

// Intra-wave LDS ordering + compiler barrier (CDNA5 split counters).
__device__ inline void lds_fence() { asm volatile("s_wait_dscnt 0" ::: "memory"); }

__device__ inline float wave_sum(float v) {
#pragma unroll
    for (int off = 16; off > 0; off >>= 1) v += __shfl_xor(v, off, 32);
    return v;
}

__device__ inline float siluf(float x) { return x / (1.f + expf(-x)); }

// D = A(16x4 f32) * B(4x16 f32) + C, full wave32 op.
__device__ inline v8f wmma4(v2f a, v2f b, v8f c) {
    return __builtin_amdgcn_wmma_f32_16x16x4_f32(false, a, false, b, (short)0, c, false, false);
}

// A fragment: 16x4 tile at columns [k0,k0+4) of row-major A (lda floats/row).
// lanes 0-15: M=lane, K=k0+0/1 ; lanes 16-31: M=lane-16, K=k0+2/3
__device__ inline v2f load_a_frag(const float* A, int lda, int k0, int lane) {
    int m  = lane & 15;
    int kk = k0 + ((lane >> 4) << 1);
    v2f a;
    a.x = A[m * lda + kk];
    a.y = A[m * lda + kk + 1];
    return a;
}

// B fragment: 4x16 tile at rows [k0,k0+4), cols [n0,n0+16) of row-major B.
// VGPR0: K=k0 (+2 for upper half), VGPR1: K=k0+1 (+2 for upper half)
__device__ inline v2f load_b_frag(const float* B, int ldb, int k0, int n0, int lane) {
    int n  = (lane & 15) + n0;
    int kk = k0 + ((lane >> 4) << 1);
    v2f b;
    b.x = B[kk * ldb + n];
    b.y = B[(kk + 1) * ldb + n];
    return b;
}

// ---------------------------------------------------------------- K0: init
__global__ void init_acc_kernel(float* inv_acc, float* equ_acc, float* nb, int n) {
    int i = blockIdx.x * blockDim.x + threadIdx.x;
    if (i < n * 128) {
        inv_acc[i] = 0.f;
        equ_acc[i] = 0.f;
    }
    if (i < n) nb[i] = 1.f;
}

// ------------------------------------------------- K1: node pre (wave/node)
__global__ void node_pre_kernel(const float* __restrict__ feats, const float* __restrict__ equ,
                                const float* __restrict__ ln1_w, const float* __restrict__ ln1_b,
                                const float* __restrict__ eqn1_w, const float* __restrict__ eqn1_b,
                                const float* __restrict__ inv_fc_w, const float* __restrict__ inv_fc_b,
                                const float* __restrict__ equ_fc_w,
                                float* __restrict__ feats_ln, float* __restrict__ equ_ln,
                                float* __restrict__ feats1, float* __restrict__ equ1, int n) {
    __shared__ float ybuf[8][128];
    __shared__ float ebuf[8][4][32];
    int wave = threadIdx.x >> 5, lane = threadIdx.x & 31;
    int node = blockIdx.x * 8 + wave;
    if (node >= n) return;
    size_t fb = (size_t)node * 128;

    // feats layer norm
    const float* x = feats + fb;
    float v[4], s = 0.f;
#pragma unroll
    for (int j = 0; j < 4; ++j) { v[j] = x[lane + 32 * j]; s += v[j]; }
    float mu = wave_sum(s) * (1.f / 128.f);
    float var = 0.f;
#pragma unroll
    for (int j = 0; j < 4; ++j) { float d0 = v[j] - mu; var += d0 * d0; }
    var = wave_sum(var) * (1.f / 128.f);
    float rs = rsqrtf(var + EPSF);
#pragma unroll
    for (int j = 0; j < 4; ++j) {
        int i = lane + 32 * j;
        float y = (v[j] - mu) * rs * ln1_w[i] + ln1_b[i];
        feats_ln[fb + i] = y;
        ybuf[wave][i] = y;
    }

    // equ layer norm (lane = channel)
    const float* eq = equ + fb;
    float e0 = eq[lane], e1 = eq[32 + lane], e2 = eq[64 + lane], e3 = eq[96 + lane];
    float mu0 = wave_sum(e0) * (1.f / 32.f);
    float f0 = e0 - mu0;
    float n0 = wave_sum(f0 * f0) * (1.f / 32.f);
    f0 = f0 * rsqrtf(n0 + EPSF) * eqn1_w[lane] + eqn1_b[lane];
    float n1 = wave_sum(e1 * e1 + e2 * e2 + e3 * e3) * (1.f / 96.f);
    float r1 = rsqrtf(n1 + EPSF) * eqn1_w[32 + lane];
    float f1 = e1 * r1, f2 = e2 * r1, f3 = e3 * r1;
    equ_ln[fb + lane] = f0; equ_ln[fb + 32 + lane] = f1;
    equ_ln[fb + 64 + lane] = f2; equ_ln[fb + 96 + lane] = f3;
    ebuf[wave][0][lane] = f0; ebuf[wave][1][lane] = f1;
    ebuf[wave][2][lane] = f2; ebuf[wave][3][lane] = f3;
    lds_fence();

    // feats1 = feats_ln @ inv_fc_w + b  (lane = out channel)
    float acc = inv_fc_b[lane];
    for (int k = 0; k < 128; ++k) acc += ybuf[wave][k] * inv_fc_w[k * 32 + lane];
    feats1[(size_t)node * 32 + lane] = acc;

    // equ1 = equ_ln @ equ_fc_w
#pragma unroll
    for (int l = 0; l < 4; ++l) {
        float a = 0.f;
        for (int k = 0; k < 32; ++k) a += ebuf[wave][l][k] * equ_fc_w[k * 32 + lane];
        equ1[fb + l * 32 + lane] = a;
    }
}

// ------------------------------------------------- K2: edge stage (WMMA f32)
#define EK_WAVES 8
#define EK_WAVE_FLOATS 2656  // Astage 2048 + l0buf 512 + shbuf 64 + idx 32
#define EK_WEIGHT_FLOATS 15520

__launch_bounds__(256, 1)
__global__ void edge_kernel(const int* __restrict__ nbr, const float* __restrict__ points,
                            const float* __restrict__ feats1, const float* __restrict__ equ1,
                            const float* __restrict__ fc_l0_w, const float* __restrict__ fc_l0_b,
                            const float* __restrict__ equ_fc_2_w,
                            const float* __restrict__ inv_fc_1_w, const float* __restrict__ inv_fc_1_b,
                            float* __restrict__ inv_acc, float* __restrict__ equ_acc,
                            float* __restrict__ nb_acc, int E, int nTiles) {
    extern __shared__ float smem[];
    float* w_l0  = smem;               // 64*32
    float* b_l0  = w_l0 + 64 * 32;     // 32
    float* w_e2  = b_l0 + 32;          // 32*32
    float* w_inv = w_e2 + 32 * 32;     // 96*128
    float* b_inv = w_inv + 96 * 128;   // 128
    float* wavebase = b_inv + 128;

    int tid = threadIdx.x;
    for (int i = tid; i < 64 * 32; i += blockDim.x) w_l0[i] = fc_l0_w[i];
    for (int i = tid; i < 32; i += blockDim.x) b_l0[i] = fc_l0_b[i];
    for (int i = tid; i < 32 * 32; i += blockDim.x) w_e2[i] = equ_fc_2_w[i];
    for (int i = tid; i < 96 * 128; i += blockDim.x) w_inv[i] = inv_fc_1_w[i];
    for (int i = tid; i < 128; i += blockDim.x) b_inv[i] = inv_fc_1_b[i];
    __syncthreads();

    int wave = tid >> 5, lane = tid & 31;
    float* Astage = wavebase + wave * EK_WAVE_FLOATS;
    float* l0buf  = Astage + 2048;   // 16x32
    float* shbuf  = l0buf + 512;     // 16x4
    int*   idxs   = (int*)(shbuf + 64); // [0..15]=src [16..31]=dst

    const int c = lane;              // channel index for gather phase
    const int n_lo = lane & 15;
    const int mbase = (lane >> 4) << 3;

    for (int tile = blockIdx.x * EK_WAVES + wave; tile < nTiles; tile += gridDim.x * EK_WAVES) {
        int e0 = tile * 16;
        if (lane < 16) {
            int e = e0 + lane;
            int s = nbr[e], d = nbr[E + e];
            idxs[lane] = s;
            idxs[16 + lane] = d;
            shbuf[lane * 4 + 0] = 1.f;
            shbuf[lane * 4 + 1] = SQRT3F * (points[3 * s + 0] - points[3 * d + 0]);
            shbuf[lane * 4 + 2] = SQRT3F * (points[3 * s + 1] - points[3 * d + 1]);
            shbuf[lane * 4 + 3] = SQRT3F * (points[3 * s + 2] - points[3 * d + 2]);
        }
        lds_fence();

        // ---- dots: A1[e][c]=dot1, A1[e][32+c]=dot2 (A1 is 16x64 in Astage)
        for (int e = 0; e < 16; ++e) {
            int s = idxs[e], d = idxs[16 + e];
            const float* se = equ1 + (size_t)s * 128;
            const float* te = equ1 + (size_t)d * 128;
            float sh1 = shbuf[e * 4 + 1], sh2 = shbuf[e * 4 + 2], sh3 = shbuf[e * 4 + 3];
            float s0 = se[c], s1 = se[32 + c], s2 = se[64 + c], s3 = se[96 + c];
            float d1 = s0 * te[c] + s1 * te[32 + c] + s2 * te[64 + c] + s3 * te[96 + c];
            float d2 = s0 + s1 * sh1 + s2 * sh2 + s3 * sh3;
            Astage[e * 64 + c] = d1;
            Astage[e * 64 + 32 + c] = d2;
        }
        lds_fence();

        // ---- GEMM1: equ_l0 = silu(A1[16x64] @ w_l0[64x32] + b_l0)
        v8f c0 = v8f_zero(), c1 = v8f_zero();
        for (int k = 0; k < 64; k += 4) {
            v2f a = load_a_frag(Astage, 64, k, lane);
            c0 = wmma4(a, load_b_frag(w_l0, 32, k, 0, lane), c0);
            c1 = wmma4(a, load_b_frag(w_l0, 32, k, 16, lane), c1);
        }
#pragma unroll
        for (int r = 0; r < 8; ++r) {
            int m = r + mbase;
            float x0 = siluf(c0[r] + b_l0[n_lo]);
            float x1 = siluf(c1[r] + b_l0[16 + n_lo]);
            l0buf[m * 32 + n_lo] = x0;
            l0buf[m * 32 + 16 + n_lo] = x1;
        }
        lds_fence();

        // ---- build A2 (64x32): row e*4+0 = equ_l0[e]; rows e*4+l = equ1[src[e]][l]
        for (int e = 0; e < 16; ++e) {
            int s = idxs[e];
            const float* se = equ1 + (size_t)s * 128;
            Astage[(e * 4 + 0) * 32 + c] = l0buf[e * 32 + c];
            Astage[(e * 4 + 1) * 32 + c] = se[32 + c];
            Astage[(e * 4 + 2) * 32 + c] = se[64 + c];
            Astage[(e * 4 + 3) * 32 + c] = se[96 + c];
        }
        lds_fence();

        // ---- GEMM2: equ_message = A2[64x32] @ w_e2[32x32], scatter to equ_acc
        for (int mt = 0; mt < 4; ++mt) {
            v8f d0 = v8f_zero(), d1 = v8f_zero();
            const float* Am = Astage + mt * 16 * 32;
            for (int k = 0; k < 32; k += 4) {
                v2f a = load_a_frag(Am, 32, k, lane);
                d0 = wmma4(a, load_b_frag(w_e2, 32, k, 0, lane), d0);
                d1 = wmma4(a, load_b_frag(w_e2, 32, k, 16, lane), d1);
            }
#pragma unroll
            for (int r = 0; r < 8; ++r) {
                int m = mt * 16 + r + mbase;
                int e = m >> 2, l = m & 3;
                int dnode = idxs[16 + e];
                float* p = equ_acc + (size_t)dnode * 128 + l * 32;
                atomicAdd(p + n_lo, d0[r]);
                atomicAdd(p + 16 + n_lo, d1[r]);
            }
        }

        // ---- build A3 (16x96): [s_feats | t_feats | equ_l0]
        for (int e = 0; e < 16; ++e) {
            int s = idxs[e], d = idxs[16 + e];
            Astage[e * 96 + c]      = feats1[(size_t)s * 32 + c];
            Astage[e * 96 + 32 + c] = feats1[(size_t)d * 32 + c];
            Astage[e * 96 + 64 + c] = l0buf[e * 32 + c];
        }
        lds_fence();

        // ---- GEMM3: inv_message = silu(A3[16x96] @ w_inv[96x128] + b_inv)
        v8f acc[8];
#pragma unroll
        for (int nt = 0; nt < 8; ++nt) acc[nt] = v8f_zero();
        for (int k = 0; k < 96; k += 4) {
            v2f a = load_a_frag(Astage, 96, k, lane);
#pragma unroll
            for (int nt = 0; nt < 8; ++nt)
                acc[nt] = wmma4(a, load_b_frag(w_inv, 128, k, nt * 16, lane), acc[nt]);
        }
#pragma unroll
        for (int r = 0; r < 8; ++r) {
            float vals[8];
            float rmax = -3.4e38f;
#pragma unroll
            for (int nt = 0; nt < 8; ++nt) {
                float x = siluf(acc[nt][r] + b_inv[nt * 16 + n_lo]);
                vals[nt] = x;
                rmax = fmaxf(rmax, x);
            }
#pragma unroll
            for (int off = 1; off < 16; off <<= 1)
                rmax = fmaxf(rmax, __shfl_xor(rmax, off, 32));
            int e = r + mbase;
            int dnode = idxs[16 + e];
            float* p = inv_acc + (size_t)dnode * 128;
#pragma unroll
            for (int nt = 0; nt < 8; ++nt) atomicAdd(p + nt * 16 + n_lo, vals[nt]);
            if (n_lo == 0) atomicAdd(nb_acc + dnode, rmax > 0.f ? 1.f : 0.f);
        }
    }
}

// ------------------------------------------------- K2b: scalar tail (E % 16)
__global__ void edge_tail_kernel(const int* __restrict__ nbr, const float* __restrict__ points,
                                 const float* __restrict__ feats1, const float* __restrict__ equ1,
                                 const float* __restrict__ fc_l0_w, const float* __restrict__ fc_l0_b,
                                 const float* __restrict__ equ_fc_2_w,
                                 const float* __restrict__ inv_fc_1_w, const float* __restrict__ inv_fc_1_b,
                                 float* __restrict__ inv_acc, float* __restrict__ equ_acc,
                                 float* __restrict__ nb_acc, int eStart, int E) {
    __shared__ float Ad[64];
    __shared__ float l0[32];
    int e = eStart + blockIdx.x;
    if (e >= E) return;
    int c = threadIdx.x & 31;
    int s = nbr[e], d = nbr[E + e];
    float sh1 = SQRT3F * (points[3 * s + 0] - points[3 * d + 0]);
    float sh2 = SQRT3F * (points[3 * s + 1] - points[3 * d + 1]);
    float sh3 = SQRT3F * (points[3 * s + 2] - points[3 * d + 2]);
    const float* se = equ1 + (size_t)s * 128;
    const float* te = equ1 + (size_t)d * 128;
    float s0 = se[c], s1 = se[32 + c], s2 = se[64 + c], s3 = se[96 + c];
    Ad[c]      = s0 * te[c] + s1 * te[32 + c] + s2 * te[64 + c] + s3 * te[96 + c];
    Ad[32 + c] = s0 + s1 * sh1 + s2 * sh2 + s3 * sh3;
    lds_fence();
    float acc = fc_l0_b[c];
    for (int k = 0; k < 64; ++k) acc += Ad[k] * fc_l0_w[k * 32 + c];
    l0[c] = siluf(acc);
    lds_fence();
    {
        float r0 = 0, r1 = 0, r2 = 0, r3 = 0;
        for (int k = 0; k < 32; ++k) {
            float w = equ_fc_2_w[k * 32 + c];
            r0 += l0[k] * w;
            r1 += se[32 + k] * w;
            r2 += se[64 + k] * w;
            r3 += se[96 + k] * w;
        }
        float* p = equ_acc + (size_t)d * 128;
        atomicAdd(p + c, r0); atomicAdd(p + 32 + c, r1);
        atomicAdd(p + 64 + c, r2); atomicAdd(p + 96 + c, r3);
    }
    float vals[4], rmax = -3.4e38f;
    for (int j = 0; j < 4; ++j) {
        int i = c + 32 * j;
        float a = inv_fc_1_b[i];
        for (int k = 0; k < 32; ++k) {
            a += feats1[(size_t)s * 32 + k] * inv_fc_1_w[k * 128 + i];
            a += feats1[(size_t)d * 32 + k] * inv_fc_1_w[(32 + k) * 128 + i];
            a += l0[k] * inv_fc_1_w[(64 + k) * 128 + i];
        }
        a = siluf(a);
        vals[j] = a;
        rmax = fmaxf(rmax, a);
    }
    for (int off = 16; off > 0; off >>= 1) rmax = fmaxf(rmax, __shfl_xor(rmax, off, 32));
    float* p = inv_acc + (size_t)d * 128;
    for (int j = 0; j < 4; ++j) atomicAdd(p + c + 32 * j, vals[j]);
    if (c == 0) atomicAdd(nb_acc + d, rmax > 0.f ? 1.f : 0.f);
}

// ------------------------------------------------- K3: inv FFN (wave/node)
__global__ void node_inv_kernel(const float* __restrict__ inv_acc, const float* __restrict__ feats_ln,
                                const float* __restrict__ nb,
                                const float* __restrict__ ffn_ln_w, const float* __restrict__ ffn_ln_b,
                                const float* __restrict__ w1, const float* __restrict__ b1,
                                const float* __restrict__ w2, const float* __restrict__ b2,
                                float* __restrict__ out_inv, int n) {
    __shared__ float hbuf[8][128];
    __shared__ float rbuf[8][128];
    int wave = threadIdx.x >> 5, lane = threadIdx.x & 31;
    int node = blockIdx.x * 8 + wave;
    if (node >= n) return;
    size_t base = (size_t)node * 128;
    float rcp = 1.f / nb[node];
    float v[4], s = 0.f;
#pragma unroll
    for (int j = 0; j < 4; ++j) {
        int i = lane + 32 * j;
        v[j] = (inv_acc[base + i] + feats_ln[base + i]) * rcp;
        s += v[j];
    }
    float mu = wave_sum(s) * (1.f / 128.f);
    float var = 0.f;
#pragma unroll
    for (int j = 0; j < 4; ++j) { float d0 = v[j] - mu; var += d0 * d0; }
    var = wave_sum(var) * (1.f / 128.f);
    float rs = rsqrtf(var + EPSF);
#pragma unroll
    for (int j = 0; j < 4; ++j) {
        int i = lane + 32 * j;
        hbuf[wave][i] = (v[j] - mu) * rs * ffn_ln_w[i] + ffn_ln_b[i];
    }
    lds_fence();
#pragma unroll
    for (int j = 0; j < 4; ++j) {
        int i = lane + 32 * j;
        float a = b1[i];
        for (int k = 0; k < 128; ++k) a += hbuf[wave][k] * w1[k * 128 + i];
        rbuf[wave][i] = fmaxf(a, 0.f);
    }
    lds_fence();
#pragma unroll
    for (int j = 0; j < 4; ++j) {
        int i = lane + 32 * j;
        float a = b2[i];
        for (int k = 0; k < 128; ++k) a += rbuf[wave][k] * w2[k * 128 + i];
        out_inv[base + i] = a + v[j];
    }
}

// ------------------------------------------------- K4: equ gate (wave/node)
__global__ void node_equ_kernel(const float* __restrict__ equ_acc, const float* __restrict__ equ_ln,
                                const float* __restrict__ nb,
                                const float* __restrict__ eqn2_w, const float* __restrict__ eqn2_b,
                                const float* __restrict__ fe1_w, const float* __restrict__ gate_w,
                                const float* __restrict__ fe2_w,
                                float* __restrict__ out_equ, int n) {
    __shared__ float gbuf[8][4][32];
    __shared__ float g0buf[8][32];
    int wave = threadIdx.x >> 5, lane = threadIdx.x & 31;
    int node = blockIdx.x * 8 + wave;
    if (node >= n) return;
    size_t base = (size_t)node * 128;
    float rcp = 1.f / nb[node];
    float q0 = (equ_acc[base + lane] + equ_ln[base + lane]) * rcp;
    float q1 = (equ_acc[base + 32 + lane] + equ_ln[base + 32 + lane]) * rcp;
    float q2 = (equ_acc[base + 64 + lane] + equ_ln[base + 64 + lane]) * rcp;
    float q3 = (equ_acc[base + 96 + lane] + equ_ln[base + 96 + lane]) * rcp;

    float mu0 = wave_sum(q0) * (1.f / 32.f);
    float f0 = q0 - mu0;
    float n0 = wave_sum(f0 * f0) * (1.f / 32.f);
    f0 = f0 * rsqrtf(n0 + EPSF) * eqn2_w[lane] + eqn2_b[lane];
    float n1 = wave_sum(q1 * q1 + q2 * q2 + q3 * q3) * (1.f / 96.f);
    float r1 = rsqrtf(n1 + EPSF) * eqn2_w[32 + lane];
    gbuf[wave][0][lane] = f0;
    gbuf[wave][1][lane] = q1 * r1;
    gbuf[wave][2][lane] = q2 * r1;
    gbuf[wave][3][lane] = q3 * r1;
    lds_fence();

    float g[4];
#pragma unroll
    for (int l = 0; l < 4; ++l) {
        float a = 0.f;
        for (int k = 0; k < 32; ++k) a += gbuf[wave][l][k] * fe1_w[k * 32 + lane];
        g[l] = a;
    }
    g0buf[wave][lane] = g[0];
    lds_fence();
    float aA = 0.f, aB = 0.f;
    for (int k = 0; k < 32; ++k) {
        float gk = g0buf[wave][k];
        aA += gk * gate_w[k * 64 + lane];
        aB += gk * gate_w[k * 64 + 32 + lane];
    }
    float mult = 1.f / (1.f + expf(-aB));
    gbuf[wave][0][lane] = siluf(aA);
    gbuf[wave][1][lane] = g[1] * mult;
    gbuf[wave][2][lane] = g[2] * mult;
    gbuf[wave][3][lane] = g[3] * mult;
    lds_fence();
    float q[4] = {q0, q1, q2, q3};
#pragma unroll
    for (int l = 0; l < 4; ++l) {
        float a = 0.f;
        for (int k = 0; k < 32; ++k) a += gbuf[wave][l][k] * fe2_w[k * 32 + lane];
        out_equ[base + l * 32 + lane] = a + q[l];
    }
}

// ---------------------------------------------------------------- launcher
extern "C" void kernel_launch(void* const* d_in, const int* in_sizes, int n_in,
                              void* d_out, int out_size, void* d_ws, size_t ws_size,
                              hipStream_t stream) {
    const float* feats      = (const float*)d_in[0];
    const float* equ        = (const float*)d_in[1];
    const float* points     = (const float*)d_in[2];
    const int*   nbr        = (const int*)d_in[3];
    const float* ln1_w      = (const float*)d_in[4];
    const float* ln1_b      = (const float*)d_in[5];
    const float* eqn1_w     = (const float*)d_in[6];
    const float* eqn1_b     = (const float*)d_in[7];
    const float* inv_fc_w   = (const float*)d_in[8];
    const float* inv_fc_b   = (const float*)d_in[9];
    const float* equ_fc_w   = (const float*)d_in[10];
    const float* fc_l0_w    = (const float*)d_in[11];
    const float* fc_l0_b    = (const float*)d_in[12];
    const float* equ_fc_2_w = (const float*)d_in[13];
    const float* inv_fc_1_w = (const float*)d_in[14];
    const float* inv_fc_1_b = (const float*)d_in[15];
    const float* ffn_ln_w   = (const float*)d_in[16];
    const float* ffn_ln_b   = (const float*)d_in[17];
    const float* ffn_w1     = (const float*)d_in[18];
    const float* ffn_b1     = (const float*)d_in[19];
    const float* ffn_w2     = (const float*)d_in[20];
    const float* ffn_b2     = (const float*)d_in[21];
    const float* eqn2_w     = (const float*)d_in[22];
    const float* eqn2_b     = (const float*)d_in[23];
    const float* fe1_w      = (const float*)d_in[24];
    const float* gate_w     = (const float*)d_in[25];
    const float* fe2_w      = (const float*)d_in[26];

    int N = in_sizes[0] / 128;
    int E = in_sizes[3] / 2;

    float* ws = (float*)d_ws;
    size_t o = 0;
    float* feats_ln = ws + o; o += (size_t)N * 128;
    float* equ_ln   = ws + o; o += (size_t)N * 128;
    float* feats1   = ws + o; o += (size_t)N * 32;
    float* equ1     = ws + o; o += (size_t)N * 128;
    float* inv_acc  = ws + o; o += (size_t)N * 128;
    float* equ_acc  = ws + o; o += (size_t)N * 128;
    float* nb       = ws + o; o += (size_t)N;

    float* out_inv = (float*)d_out;
    float* out_equ = out_inv + (size_t)N * 128;

    init_acc_kernel<<<(N * 128 + 255) / 256, 256, 0, stream>>>(inv_acc, equ_acc, nb, N);

    int nodeBlocks = (N + 7) / 8;
    node_pre_kernel<<<nodeBlocks, 256, 0, stream>>>(feats, equ, ln1_w, ln1_b, eqn1_w, eqn1_b,
                                                    inv_fc_w, inv_fc_b, equ_fc_w,
                                                    feats_ln, equ_ln, feats1, equ1, N);

    int nTiles = E / 16;
    int rem = E - nTiles * 16;
    if (nTiles > 0) {
        int grid = (nTiles + EK_WAVES - 1) / EK_WAVES;
        if (grid > 2048) grid = 2048;
        size_t smemBytes = (size_t)(EK_WEIGHT_FLOATS + EK_WAVES * EK_WAVE_FLOATS) * sizeof(float);
        edge_kernel<<<grid, 256, smemBytes, stream>>>(nbr, points, feats1, equ1,
                                                      fc_l0_w, fc_l0_b, equ_fc_2_w,
                                                      inv_fc_1_w, inv_fc_1_b,
                                                      inv_acc, equ_acc, nb, E, nTiles);
    }
    if (rem > 0) {
        edge_tail_kernel<<<rem, 32, 0, stream>>>(nbr, points, feats1, equ1,
                                                 fc_l0_w, fc_l0_b, equ_fc_2_w,
                                                 inv_fc_1_w, inv_fc_1_b,
                                                 inv_acc, equ_acc, nb, nTiles * 16, E);
    }

    node_inv_kernel<<<nodeBlocks, 256, 0, stream>>>(inv_acc, feats_ln, nb, ffn_ln_w, ffn_ln_b,
                                                    ffn_w1, ffn_b1, ffn_w2, ffn_b2, out_inv, N);
    node_equ_kernel<<<nodeBlocks, 256, 0, stream>>>(equ_acc, equ_ln, nb, eqn2_w, eqn2_b,
                                                    fe1_w, gate_w, fe2_w, out_equ, N);
}